// Encoder_34394098106409
// MI455X (gfx1250) — compile-verified
//
#include <hip/hip_runtime.h>
#include <math.h>

// ---------------- problem constants ----------------
#define NENT   50000
#define NRELS  500
#define DIM    256
#define HALFD  128
#define EDIR   250000
#define E2     (2*EDIR)
#define ROT_SCALE 2.0943951023931953f   // PI / 1.5

// ws layout (float offsets)
#define WS_AGG_IN   0
#define WS_AGG_OUT  12800000
#define WS_DEG_IN   25600000
#define WS_DEG_OUT  25650000
#define WS_WEFF     25700000
#define WS_ZERO_N   25700000   // floats to zero (aggs + degs)

typedef float v2f __attribute__((ext_vector_type(2)));
typedef float v8f __attribute__((ext_vector_type(8)));

// ---------------- zero scratch ----------------
__global__ void k_zero(float* __restrict__ p, int n) {
    int i = blockIdx.x * blockDim.x + threadIdx.x;
    if (i < n) p[i] = 0.0f;
}

// ---------------- w_eff = R @ w_loop (fold self-loop rotation into W) ------
__global__ void k_weff(const float* __restrict__ w_loop,
                       const float* __restrict__ loop_rel,
                       float* __restrict__ w_eff) {
    int i = blockIdx.x * blockDim.x + threadIdx.x;
    if (i >= DIM * DIM) return;
    int m = i >> 8, n = i & 255;
    int mm = m & (HALFD - 1);
    float r = loop_rel[mm] * ROT_SCALE;
    float s, c;
    __sincosf(r, &s, &c);
    float a = w_loop[mm * DIM + n];
    float b = w_loop[(mm + HALFD) * DIM + n];
    w_eff[i] = (m < HALFD) ? (c * a + s * b) : (s * a - c * b);
}

// ---------------- per-direction degree (over destination rows) -------------
__global__ void k_deg(const int* __restrict__ eidx,
                      float* __restrict__ deg_in, float* __restrict__ deg_out) {
    int i = blockIdx.x * blockDim.x + threadIdx.x;
    if (i >= E2) return;
    int row = eidx[i];                       // edge_index[0][i]
    float* deg = (i < EDIR) ? deg_in : deg_out;
    atomicAdd(&deg[row], 1.0f);
}

__global__ void k_dinv(float* __restrict__ deg, int n) {
    int i = blockIdx.x * blockDim.x + threadIdx.x;
    if (i >= n) return;
    float d = deg[i];
    deg[i] = (d > 0.0f) ? rsqrtf(d) : 0.0f;
}

// ---------------- per-edge rotate + norm-scale + scatter-add ---------------
// One wave32 per edge; lane j owns 4 elems of each 128-wide half (float4).
__global__ void k_scatter(const float* __restrict__ x,
                          const int* __restrict__ eidx,
                          const int* __restrict__ etype,
                          const float* __restrict__ rel,
                          const float* __restrict__ dinv_in,
                          const float* __restrict__ dinv_out,
                          float* __restrict__ agg_in,
                          float* __restrict__ agg_out) {
    int wave = (blockIdx.x * blockDim.x + threadIdx.x) >> 5;
    int lane = threadIdx.x & 31;
    if (wave >= E2) return;
    int row = eidx[wave];
    int col = eidx[E2 + wave];
    int et  = etype[wave];
    bool is_in = (wave < EDIR);
    const float* dinv = is_in ? dinv_in : dinv_out;
    float*       agg  = is_in ? agg_in  : agg_out;
    float nrm = dinv[row] * dinv[col];

    int off = lane * 4;
    const float4 re = *(const float4*)(x + (size_t)col * DIM + off);
    const float4 im = *(const float4*)(x + (size_t)col * DIM + HALFD + off);
    const float4 rr = *(const float4*)(rel + (size_t)et * HALFD + off);

    float s0,c0,s1,c1,s2,c2,s3,c3;
    __sincosf(rr.x * ROT_SCALE, &s0, &c0);
    __sincosf(rr.y * ROT_SCALE, &s1, &c1);
    __sincosf(rr.z * ROT_SCALE, &s2, &c2);
    __sincosf(rr.w * ROT_SCALE, &s3, &c3);

    float* dst = agg + (size_t)row * DIM;
    atomicAdd(dst + off + 0,          (re.x * c0 + im.x * s0) * nrm);
    atomicAdd(dst + off + 1,          (re.y * c1 + im.y * s1) * nrm);
    atomicAdd(dst + off + 2,          (re.z * c2 + im.z * s2) * nrm);
    atomicAdd(dst + off + 3,          (re.w * c3 + im.w * s3) * nrm);
    atomicAdd(dst + HALFD + off + 0,  (re.x * s0 - im.x * c0) * nrm);
    atomicAdd(dst + HALFD + off + 1,  (re.y * s1 - im.y * c1) * nrm);
    atomicAdd(dst + HALFD + off + 2,  (re.z * s2 - im.z * c2) * nrm);
    atomicAdd(dst + HALFD + off + 3,  (re.w * s3 - im.w * c3) * nrm);
}

// ---------------- WMMA f32 16x16x4 B-fragment helper -----------------------
__device__ __forceinline__ v2f bfrag(const float* __restrict__ B, int ldb,
                                     int bcol, int ka) {
    v2f b;
    b[0] = B[(size_t)ka * ldb + bcol];
    b[1] = B[(size_t)(ka + 1) * ldb + bcol];
    return b;
}

// ---------------- fused triple GEMM:  out = (Ain@Win + Aout@Wout + x@Weff)/3
// One block per 16-row M-panel (3125 blocks, 8 waves). A K-slabs staged in
// LDS (pad 68 -> conflict-free ds_load_b64); each wave owns two 16x16 N-tiles
// so every A fragment feeds two WMMAs.
#define KSLAB 64
#define APAD  68
__global__ void k_gemm_main(const float* __restrict__ agg_in,
                            const float* __restrict__ agg_out,
                            const float* __restrict__ x,
                            const float* __restrict__ w_in,
                            const float* __restrict__ w_out,
                            const float* __restrict__ w_eff,
                            float* __restrict__ out) {
    __shared__ float lA[3][16][APAD];

    const int mt   = blockIdx.x;             // 0..3124
    const int tid  = threadIdx.x;
    const int wv   = tid >> 5;               // 0..7
    const int lane = tid & 31;
    const int half = lane >> 4;
    const int l15  = lane & 15;

    const float* Amat[3] = { agg_in, agg_out, x };
    const float* Bmat[3] = { w_in,   w_out,   w_eff };

    const int bcol0 = (2 * wv) * 16 + l15;   // first N-tile column
    const int bcol1 = bcol0 + 16;            // second N-tile column

    // cooperative-load coords: thread t loads row r, 4 cols at cs
    const int r  = tid >> 4;                 // 0..15
    const int cs = (tid & 15) * 4;           // 0..60

    v8f acc0 = {};
    v8f acc1 = {};

    for (int k0 = 0; k0 < DIM; k0 += KSLAB) {
        // stage 16 x 64 K-slab of all three A matrices into LDS
#pragma unroll
        for (int p = 0; p < 3; ++p) {
            const float4 v = *(const float4*)(Amat[p] +
                (size_t)(mt * 16 + r) * DIM + k0 + cs);
            *(float4*)&lA[p][r][cs] = v;
        }
        __syncthreads();

#pragma unroll 4
        for (int k = 0; k < KSLAB; k += 4) {
            const int ka = k0 + k + half * 2;      // global K for B
            const int kl = k + half * 2;           // local K for A slab
#pragma unroll
            for (int p = 0; p < 3; ++p) {
                v2f a = *(const v2f*)&lA[p][l15][kl];
                acc0 = __builtin_amdgcn_wmma_f32_16x16x4_f32(
                    false, a, false, bfrag(Bmat[p], DIM, bcol0, ka),
                    (short)0, acc0, false, false);
                acc1 = __builtin_amdgcn_wmma_f32_16x16x4_f32(
                    false, a, false, bfrag(Bmat[p], DIM, bcol1, ka),
                    (short)0, acc1, false, false);
            }
        }
        __syncthreads();
    }

    const float third = 1.0f / 3.0f;
#pragma unroll
    for (int q = 0; q < 8; ++q) {
        const int orow = mt * 16 + half * 8 + q;   // C layout: VGPR q -> M=q / M=8+q
        out[(size_t)orow * DIM + bcol0] = acc0[q] * third;
        out[(size_t)orow * DIM + bcol1] = acc1[q] * third;
    }
}

// ---------------- rel_out = rel_embed(500x128) @ w_rel(128x128) ------------
__global__ void k_gemm_rel(const float* __restrict__ rel_embed,
                           const float* __restrict__ w_rel,
                           float* __restrict__ rel_out) {
    int wave = (blockIdx.x * blockDim.x + threadIdx.x) >> 5;
    int lane = threadIdx.x & 31;
    int nt = wave & 7;
    int mt = wave >> 3;
    if (mt >= 32) return;
    int half = lane >> 4;
    int l15  = lane & 15;
    int arow = mt * 16 + l15;
    if (arow >= NRELS) arow = NRELS - 1;   // clamp (junk rows masked at store)
    int bcol = nt * 16 + l15;

    v8f acc = {};
#pragma unroll 4
    for (int k = 0; k < HALFD; k += 4) {
        int ka = k + half * 2;
        v2f a = *(const v2f*)(rel_embed + (size_t)arow * HALFD + ka);
        acc = __builtin_amdgcn_wmma_f32_16x16x4_f32(
            false, a, false, bfrag(w_rel, HALFD, bcol, ka),
            (short)0, acc, false, false);
    }

#pragma unroll
    for (int q = 0; q < 8; ++q) {
        int orow = mt * 16 + half * 8 + q;
        if (orow < NRELS) rel_out[(size_t)orow * HALFD + bcol] = acc[q];
    }
}

// ---------------- launch ----------------------------------------------------
extern "C" void kernel_launch(void* const* d_in, const int* in_sizes, int n_in,
                              void* d_out, int out_size, void* d_ws, size_t ws_size,
                              hipStream_t stream) {
    const float* x        = (const float*)d_in[0];
    const int*   eidx     = (const int*)  d_in[1];   // (2, 500000) row-major
    const int*   etype    = (const int*)  d_in[2];
    const float* rel      = (const float*)d_in[3];   // (500,128)
    const float* w_loop   = (const float*)d_in[4];
    const float* w_in     = (const float*)d_in[5];
    const float* w_out    = (const float*)d_in[6];
    const float* w_rel    = (const float*)d_in[7];
    const float* loop_rel = (const float*)d_in[8];   // (1,128)

    float* ws       = (float*)d_ws;
    float* agg_in   = ws + WS_AGG_IN;
    float* agg_out  = ws + WS_AGG_OUT;
    float* dinv_in  = ws + WS_DEG_IN;
    float* dinv_out = ws + WS_DEG_OUT;
    float* w_eff    = ws + WS_WEFF;

    float* out     = (float*)d_out;
    float* rel_out = out + (size_t)NENT * DIM;       // outputs concatenated

    // 1) zero accumulators + degree counters
    k_zero<<<(WS_ZERO_N + 255) / 256, 256, 0, stream>>>(ws, WS_ZERO_N);
    // 2) fold self-loop rotation into w_loop
    k_weff<<<(DIM * DIM + 255) / 256, 256, 0, stream>>>(w_loop, loop_rel, w_eff);
    // 3) degrees per direction, then deg^{-1/2}
    k_deg<<<(E2 + 255) / 256, 256, 0, stream>>>(eidx, dinv_in, dinv_out);
    k_dinv<<<(2 * NENT + 255) / 256, 256, 0, stream>>>(dinv_in, 2 * NENT);
    // 4) rotate + scale + scatter-add messages (1 wave per edge)
    k_scatter<<<(E2 * 32 + 255) / 256, 256, 0, stream>>>(
        x, eidx, etype, rel, dinv_in, dinv_out, agg_in, agg_out);
    // 5) fused triple GEMM -> out (one block per 16-row panel)
    k_gemm_main<<<NENT / 16, 256, 0, stream>>>(
        agg_in, agg_out, x, w_in, w_out, w_eff, out);
    // 6) rel_out GEMM
    k_gemm_rel<<<32, 256, 0, stream>>>(rel, w_rel, rel_out);
}